// IntrinsicLoss_9294309228753
// MI455X (gfx1250) — compile-verified
//
#include <hip/hip_runtime.h>
#include <hip/hip_bf16.h>

// IntrinsicLoss: out[b] = || A_b * w_b ||^2 with A_b (3x4) built from X[b,0:6].
//
// Roofline on MI455X: 44 bytes + ~40 f32 FLOPs per element -> AI ~ 0.9.
// Pure HBM-bandwidth-bound (ideal ~4 us @ 23.3 TB/s for 92 MB of traffic).
// WMMA is NOT applicable: every batch element has a private 3x4 A and private
// w (batched independent tiny quadratic forms) -- there is no shared operand
// to tile into a 16x16 wave-cooperative matmul. The optimal CDNA5 path is
// wide coalesced global_load_b128 + v_fma_f32 chains + b64 stores, plus
// gfx1250 global_prefetch_b8 for the streaming access pattern.
//
// Layout trick: process 2 elements per thread so X (2 rows = 12 floats = 48 B)
// is exactly 3x float4 (16B-aligned: pair p starts at byte 48*p), W is 2x
// float4, and the output is one aligned float2 store.

__device__ __forceinline__ float loss_one(float x0, float x1, float x2,
                                          float x3, float x4, float x5,
                                          float4 w) {
    // row0 = [x0*x2 + x1*x3, x0+x2, x1+x3, 1]
    // row1 = [x0*x4 + x1*x5, x0+x4, x1+x5, 1]
    // row2 = [x2*x4 + x3*x5, x2+x4, x3+x5, 1]
    float r0 = fmaf(fmaf(x0, x2, x1 * x3), w.x,
                    fmaf(x0 + x2, w.y, fmaf(x1 + x3, w.z, w.w)));
    float r1 = fmaf(fmaf(x0, x4, x1 * x5), w.x,
                    fmaf(x0 + x4, w.y, fmaf(x1 + x5, w.z, w.w)));
    float r2 = fmaf(fmaf(x2, x4, x3 * x5), w.x,
                    fmaf(x2 + x4, w.y, fmaf(x3 + x5, w.z, w.w)));
    return fmaf(r0, r0, fmaf(r1, r1, r2 * r2));
}

__global__ __launch_bounds__(256) void intrinsic_loss_kernel(
    const float* __restrict__ X, const float* __restrict__ W,
    float* __restrict__ out, int B) {
    const int nPairs = B >> 1;
    const int p = blockIdx.x * blockDim.x + threadIdx.x;

    if (p < nPairs) {
        const float4* __restrict__ Xv =
            reinterpret_cast<const float4*>(X + (size_t)p * 12);
        const float4* __restrict__ Wv =
            reinterpret_cast<const float4*>(W + (size_t)p * 8);

        // gfx1250 streaming prefetch (global_prefetch_b8): speculative,
        // OOB-safe per ISA 10.5; pull a few KB ahead of the read stream.
        __builtin_prefetch(X + (size_t)p * 12 + 4096, 0, 1);
        __builtin_prefetch(W + (size_t)p * 8 + 4096, 0, 1);

        const float4 a = Xv[0];   // elem 2p:   x0 x1 x2 x3
        const float4 b = Xv[1];   // elem 2p:   x4 x5 | elem 2p+1: x0 x1
        const float4 c = Xv[2];   // elem 2p+1: x2 x3 x4 x5
        const float4 w0 = Wv[0];
        const float4 w1 = Wv[1];

        const float l0 = loss_one(a.x, a.y, a.z, a.w, b.x, b.y, w0);
        const float l1 = loss_one(b.z, b.w, c.x, c.y, c.z, c.w, w1);

        *reinterpret_cast<float2*>(out + (size_t)p * 2) = make_float2(l0, l1);
    }

    // Tail element if B is odd (not hit for B = 2^21, kept for robustness).
    if ((B & 1) && p == 0) {
        const int i = B - 1;
        const float* xr = X + (size_t)i * 6;
        const float4 w = *reinterpret_cast<const float4*>(W + (size_t)i * 4);
        out[i] = loss_one(xr[0], xr[1], xr[2], xr[3], xr[4], xr[5], w);
    }
}

extern "C" void kernel_launch(void* const* d_in, const int* in_sizes, int n_in,
                              void* d_out, int out_size, void* d_ws,
                              size_t ws_size, hipStream_t stream) {
    const float* X = (const float*)d_in[0];  // [B, 6] f32
    const float* W = (const float*)d_in[1];  // [B, 4] f32
    float* out = (float*)d_out;              // [B] f32

    const int B = in_sizes[0] / 6;
    const int nPairs = B >> 1;
    const int work = nPairs > 0 ? nPairs : 1;  // ensure tail thread exists

    const dim3 block(256);
    const dim3 grid((work + block.x - 1) / block.x);
    intrinsic_loss_kernel<<<grid, block, 0, stream>>>(X, W, out, B);
}